// BPNN_Elec_Net_56659208568936
// MI455X (gfx1250) — compile-verified
//
#include <hip/hip_runtime.h>
#include <math.h>

#define NATOMS 384
#define NSPLIT 8            // j-range splits for the angular kernel
#define JCHUNK (NATOMS / NSPLIT)
#define ETA_C   0.5f
#define RCUT_C  6.0f
#define ZETA_C  0.6f
#define LAMDA_C 0.8f
#define RS_C    1.0f
#define PI_F 3.14159265358979323846f
// 2^(1-zeta) * 0.5  (0.5 from the j<k -> full-sum symmetry)
#define ANG_SCALE 0.6597539553864471f

typedef float v2f __attribute__((ext_vector_type(2)));
typedef float v8f __attribute__((ext_vector_type(8)));

// ---------------------------------------------------------------------------
// Kernel 1: pairwise quantities. One block per atom i.
//   Amat[i][j]  = exp(-eta*D2)*fc(D)   (0 on diagonal / beyond cutoff)
//   invD[i][j]  = 1/D                  (0 on diagonal)
//   Rad[i]      = radial symmetry function
// ---------------------------------------------------------------------------
__global__ void k_pairwise(const float* __restrict__ x, float* __restrict__ Amat,
                           float* __restrict__ invD, float* __restrict__ Rad) {
  const int i = blockIdx.x;
  const int t = threadIdx.x;
  __shared__ float red[128];
  const float xix = x[3 * i + 0], xiy = x[3 * i + 1], xiz = x[3 * i + 2];
  float racc = 0.f;
  for (int j = t; j < NATOMS; j += 128) {
    float dx = xix - x[3 * j + 0];
    float dy = xiy - x[3 * j + 1];
    float dz = xiz - x[3 * j + 2];
    float d2 = dx * dx + dy * dy + dz * dz;
    float a = 0.f, id = 0.f;
    if (j != i) {
      float d = sqrtf(d2);
      id = 1.f / d;
      if (d <= RCUT_C) {
        float fc = 0.5f * (cosf(PI_F * d / RCUT_C) + 1.f);
        a = expf(-ETA_C * d2) * fc;
        float dr = d - RS_C;
        racc += expf(-ETA_C * dr * dr) * fc;
      }
    }
    Amat[i * NATOMS + j] = a;
    invD[i * NATOMS + j] = id;
  }
  red[t] = racc;
  __syncthreads();
  for (int s = 64; s > 0; s >>= 1) {
    if (t < s) red[t] += red[t + s];
    __syncthreads();
  }
  if (t == 0) Rad[i] = red[0];
}

// ---------------------------------------------------------------------------
// Kernel 2: angular symmetry function, split over j-chunks for occupancy.
// grid = (NATOMS, NSPLIT), block = 384 threads (k == tid, no inner loop).
// Summand is symmetric in (j,k) and zero on all diagonals:
//   sum_{j<k} S = 0.5 * sum_{j,k} S      (0.5 folded into ANG_SCALE).
// Deterministic: each block writes its partial to P[i][split]; no atomics.
// pow(base, 0.6) -> v_log_f32 + v_exp_f32 (base clamped to [1e-6, 1.8]).
// ---------------------------------------------------------------------------
__global__ void k_angular(const float* __restrict__ x, const float* __restrict__ Amat,
                          const float* __restrict__ invD, float* __restrict__ P) {
  const int i = blockIdx.x;
  const int split = blockIdx.y;
  const int t = threadIdx.x;          // 0..383 == k
  __shared__ float sxx[NATOMS], sxy[NATOMS], sxz[NATOMS];
  __shared__ float sA[NATOMS], sID[NATOMS];
  __shared__ float red[NATOMS];
  sxx[t] = x[3 * t + 0];
  sxy[t] = x[3 * t + 1];
  sxz[t] = x[3 * t + 2];
  sA[t] = Amat[i * NATOMS + t];
  sID[t] = invD[i * NATOMS + t];
  __syncthreads();
  const float xix = sxx[i], xiy = sxy[i], xiz = sxz[i];
  const int k = t;
  const float aik = sA[k];
  const float idik = sID[k];
  const float rkix = sxx[k] - xix;
  const float rkiy = sxy[k] - xiy;
  const float rkiz = sxz[k] - xiz;
  float acc = 0.f;
  const int j0 = split * JCHUNK;
  for (int j = j0; j < j0 + JCHUNK; ++j) {
    const float aij = sA[j];          // uniform across block
    if (aij == 0.f) continue;         // uniform skip (j==i or beyond cutoff)
    const float rijx = xix - sxx[j];
    const float rijy = xiy - sxy[j];
    const float rijz = xiz - sxz[j];
    const float idij = sID[j];
    const float ajk = Amat[(size_t)j * NATOMS + k];
    const float t2 = aij * ajk * aik;
    if (t2 != 0.f) {
      const float dotv = rijx * rkix + rijy * rkiy + rijz * rkiz;
      const float cosv = dotv * idij * idik;
      const float base = fmaxf(1.f + LAMDA_C * cosv, 1e-6f);
      // base^zeta = 2^(zeta * log2(base)) via v_log_f32 / v_exp_f32
      acc += __builtin_amdgcn_exp2f(ZETA_C * __builtin_amdgcn_logf(base)) * t2;
    }
  }
  red[t] = acc;
  __syncthreads();
  if (t < 128) red[t] += red[t + 128] + red[t + 256];
  __syncthreads();
  for (int s = 64; s > 0; s >>= 1) {
    if (t < s) red[t] += red[t + s];
    __syncthreads();
  }
  if (t == 0) P[i * NSPLIT + split] = red[0];
}

// ---------------------------------------------------------------------------
// Kernel 3: MLP via V_WMMA_F32_16X16X4_F32. One wave per 16-row tile.
// B fragment: lane<16 -> (K=k0,k0+1, N=lane); lane>=16 -> (K=k0+2,k0+3).
// W is [fout, fin] row-major; B[k][n] = W[n*fin + k]. Branchless OOB -> 0.
// ---------------------------------------------------------------------------
__device__ __forceinline__ v2f load_b_frag(const float* __restrict__ W, int fin, int fout,
                                           int kk, int nt, int row, int half) {
  const int n = nt * 16 + row;
  const int k0 = kk * 4 + half * 2;
  const int nc = (n < fout) ? n : (fout - 1);
  const int kc0 = (k0 + 0 < fin) ? (k0 + 0) : (fin - 1);
  const int kc1 = (k0 + 1 < fin) ? (k0 + 1) : (fin - 1);
  const float m0 = (n < fout && (k0 + 0) < fin) ? 1.f : 0.f;
  const float m1 = (n < fout && (k0 + 1) < fin) ? 1.f : 0.f;
  v2f b;
  b.x = W[nc * fin + kc0] * m0;
  b.y = W[nc * fin + kc1] * m1;
  return b;
}

__global__ void k_mlp(const float* __restrict__ Rad, const float* __restrict__ P,
                      const float* w1a, const float* b1a, const float* w2a,
                      const float* b2a, const float* w3a, const float* b3a,
                      const float* w1b, const float* b1b, const float* w2b,
                      const float* b2b, const float* w3b, const float* b3b,
                      float* __restrict__ out) {
  __shared__ float sh[2][16][48];  // hidden activations (16 rows x 40 cols, padded)
  __shared__ float q[2][16];
  const int m = blockIdx.x;
  const int lane = threadIdx.x;         // 0..31, one wave
  const int row = lane & 15;
  const int half = lane >> 4;
  const int grow = m * 16 + row;

  // Deterministic fixed-order reduction of the angular partials.
  float ang = 0.f;
  for (int y = 0; y < NSPLIT; ++y) ang += P[grow * NSPLIT + y];
  ang *= ANG_SCALE;
  const float rad = Rad[grow];

  for (int net = 0; net < 2; ++net) {
    const float* w1 = net ? w1b : w1a;
    const float* b1 = net ? b1b : b1a;
    const float* w2 = net ? w2b : w2a;
    const float* b2 = net ? b2b : b2a;
    const float* w3 = net ? w3b : w3a;
    const float* b3 = net ? b3b : b3a;

    // ---- layer 1: [16,2] @ [2,40]  (K padded 2->4) ----
    v2f a1;
    a1.x = (half == 0) ? ang : 0.f;  // K=0 (ang) / K=2 (pad)
    a1.y = (half == 0) ? rad : 0.f;  // K=1 (rad) / K=3 (pad)
    for (int nt = 0; nt < 3; ++nt) {
      v2f b = load_b_frag(w1, 2, 40, 0, nt, row, half);
      v8f c = {};
      c = __builtin_amdgcn_wmma_f32_16x16x4_f32(false, a1, false, b,
                                                (short)0, c, false, false);
      const int col = nt * 16 + row;
      if (col < 40) {
        const float bias = b1[col];
        for (int r = 0; r < 8; ++r) {
          const float v = c[r] + bias;
          sh[0][r + half * 8][col] = 1.f / (1.f + v * v);
        }
      }
    }
    __syncthreads();

    // ---- layer 2: [16,40] @ [40,40], 10 k-steps accumulated in C ----
    for (int nt = 0; nt < 3; ++nt) {
      v8f c = {};
      for (int kk = 0; kk < 10; ++kk) {
        const int k0 = kk * 4 + half * 2;
        v2f aa;
        aa.x = sh[0][row][k0 + 0];
        aa.y = sh[0][row][k0 + 1];
        v2f b = load_b_frag(w2, 40, 40, kk, nt, row, half);
        c = __builtin_amdgcn_wmma_f32_16x16x4_f32(false, aa, false, b,
                                                  (short)0, c, false, false);
      }
      const int col = nt * 16 + row;
      if (col < 40) {
        const float bias = b2[col];
        for (int r = 0; r < 8; ++r) {
          const float v = c[r] + bias;
          sh[1][r + half * 8][col] = 1.f / (1.f + v * v);
        }
      }
    }
    __syncthreads();

    // ---- layer 3: [16,40] @ [40,1] — VALU dot, lanes 0..15 ----
    if (half == 0) {
      float s = b3[0];
      for (int k = 0; k < 40; ++k) s += sh[1][row][k] * w3[k];
      q[net][row] = s;
    }
    __syncthreads();
  }

  if (half == 0) out[grow] = (grow == 8) ? q[1][row] : q[0][row];
}

// ---------------------------------------------------------------------------
extern "C" void kernel_launch(void* const* d_in, const int* in_sizes, int n_in,
                              void* d_out, int out_size, void* d_ws, size_t ws_size,
                              hipStream_t stream) {
  const float* x   = (const float*)d_in[0];
  const float* w1a = (const float*)d_in[1];
  const float* b1a = (const float*)d_in[2];
  const float* w2a = (const float*)d_in[3];
  const float* b2a = (const float*)d_in[4];
  const float* w3a = (const float*)d_in[5];
  const float* b3a = (const float*)d_in[6];
  const float* w1b = (const float*)d_in[7];
  const float* b1b = (const float*)d_in[8];
  const float* w2b = (const float*)d_in[9];
  const float* b2b = (const float*)d_in[10];
  const float* w3b = (const float*)d_in[11];
  const float* b3b = (const float*)d_in[12];

  float* Amat = (float*)d_ws;                  // 384*384 floats
  float* invD = Amat + NATOMS * NATOMS;        // 384*384 floats
  float* Rad  = invD + NATOMS * NATOMS;        // 384 floats
  float* P    = Rad + NATOMS;                  // 384*NSPLIT floats
  float* out  = (float*)d_out;

  k_pairwise<<<NATOMS, 128, 0, stream>>>(x, Amat, invD, Rad);
  k_angular<<<dim3(NATOMS, NSPLIT), NATOMS, 0, stream>>>(x, Amat, invD, P);
  k_mlp<<<NATOMS / 16, 32, 0, stream>>>(Rad, P, w1a, b1a, w2a, b2a, w3a, b3a,
                                        w1b, b1b, w2b, b2b, w3b, b3b, out);
}